// CompGCN_9122510537175
// MI455X (gfx1250) — compile-verified
//
#include <hip/hip_runtime.h>
#include <hip/hip_bf16.h>
#include <math.h>

#define N_ENT   100000
#define N_RELS  200
#define DIM     256
#define N_EDGE  1000000
#define HALF_E  (N_EDGE / 2)
#define BATCH   4096
#define REL_ROWS 401          // 2*N_RELS + 1
#define REL_OUT_ROWS 400

#define BM 160                // block M tile (100000 = 625 * 160)
#define BN 128                // block N tile (2 blocks cover DIM)
#define NWAVES 10             // 320 threads
#define NT_STEPS 48           // 3 sources * (256/16) K-tiles

typedef float v2f __attribute__((ext_vector_type(2)));
typedef float v8f __attribute__((ext_vector_type(8)));
typedef int   v4i __attribute__((ext_vector_type(4)));

typedef __attribute__((address_space(1))) v4i gv4i;   // global
typedef __attribute__((address_space(3))) v4i lv4i;   // LDS

#if defined(__gfx1250__) && __has_builtin(__builtin_amdgcn_global_load_async_to_lds_b128)
#define HAVE_ASYNC_LDS 1
#else
#define HAVE_ASYNC_LDS 0
#endif

#if HAVE_ASYNC_LDS
#define ASYNC_LD_B128(gp, lp)                                              \
    __builtin_amdgcn_global_load_async_to_lds_b128(                        \
        (gv4i*)(gp), (lv4i*)(lp), 0, 0)
# if __has_builtin(__builtin_amdgcn_s_wait_asynccnt)
#  define WAIT_ASYNC(n) __builtin_amdgcn_s_wait_asynccnt(n)
# else
#  define WAIT_ASYNC(n) asm volatile("s_wait_asynccnt %0" :: "i"(n) : "memory")
# endif
#else
#define WAIT_ASYNC(n) ((void)0)
#endif

__device__ __forceinline__ float dinv_of(float deg) {
    return deg > 0.0f ? rsqrtf(deg) : 0.0f;
}

__device__ __forceinline__ void atomic_fadd(float* p, float v) {
    __hip_atomic_fetch_add(p, v, __ATOMIC_RELAXED, __HIP_MEMORY_SCOPE_AGENT);
}

// ---------------------------------------------------------------------------
// 1) degree accumulation
// ---------------------------------------------------------------------------
__global__ void deg_kernel(const int* __restrict__ edge_index,
                           float* __restrict__ deg_in,
                           float* __restrict__ deg_out) {
    int e = blockIdx.x * blockDim.x + threadIdx.x;
    if (e >= N_EDGE) return;
    int row = edge_index[e];
    if (e < HALF_E) atomic_fadd(&deg_in[row], 1.0f);
    else            atomic_fadd(&deg_out[row], 1.0f);
}

// ---------------------------------------------------------------------------
// 2) rel_embed = concat(init_rel, -init_rel, loop_rel)   (401 x 256)
// ---------------------------------------------------------------------------
__global__ void build_rel_kernel(const float* __restrict__ init_rel,
                                 const float* __restrict__ loop_rel,
                                 float* __restrict__ rel_embed) {
    int idx = blockIdx.x * blockDim.x + threadIdx.x;
    if (idx >= REL_ROWS * DIM) return;
    int i = idx / DIM, d = idx % DIM;
    float v;
    if (i < N_RELS)            v =  init_rel[i * DIM + d];
    else if (i < 2 * N_RELS)   v = -init_rel[(i - N_RELS) * DIM + d];
    else                       v =  loop_rel[d];
    rel_embed[idx] = v;
}

// ---------------------------------------------------------------------------
// 3) loopW[n] = sum_k loop_rel[k] * w_loop[k][n]   (constant 1x256 row fold)
// ---------------------------------------------------------------------------
__global__ void loopw_kernel(const float* __restrict__ loopRel,
                             const float* __restrict__ wLoop,
                             float* __restrict__ loopW) {
    int n = blockIdx.x * blockDim.x + threadIdx.x;
    if (n >= DIM) return;
    float s = 0.0f;
    for (int k = 0; k < DIM; ++k) s += loopRel[k] * wLoop[k * DIM + n];
    loopW[n] = s;
}

// ---------------------------------------------------------------------------
// 4) edge aggregation: acc[row] += (embed[col] - rel_embed[etype]) * norm
// ---------------------------------------------------------------------------
__global__ void agg_kernel(const int* __restrict__ edge_index,
                           const int* __restrict__ edge_type,
                           const float* __restrict__ embed,
                           const float* __restrict__ rel_embed,
                           const float* __restrict__ deg,
                           float* __restrict__ acc,
                           int base) {
    int gid  = blockIdx.x * blockDim.x + threadIdx.x;
    int wave = gid >> 5;
    int lane = gid & 31;
    if (wave >= HALF_E) return;
    int e   = base + wave;
    int row = edge_index[e];
    int col = edge_index[N_EDGE + e];
    int rt  = edge_type[e];
    float norm = dinv_of(deg[row]) * dinv_of(deg[col]);
    const float* xr = embed     + (size_t)col * DIM;
    const float* rr = rel_embed + (size_t)rt  * DIM;
    float*       ar = acc       + (size_t)row * DIM;
#pragma unroll
    for (int j = 0; j < 8; ++j) {
        int d = lane + 32 * j;
        atomic_fadd(&ar[d], (xr[d] - rr[d]) * norm);
    }
}

// ---------------------------------------------------------------------------
// 5) block-cooperative fused triple GEMM + BN/tanh epilogue:
//    C = accIn@Win + accOut@Wout + E@Wloop        (K = 3*256, tiled by 16)
//    x = tanh(((C - loopW)/3 + bias) * gamma/sqrt(1+eps) + beta)
//    block = 160(M) x 128(N), 10 waves, double-buffered LDS tiles fed by
//    async global->LDS (gfx1250) with s_wait_asynccnt pipelining.
// ---------------------------------------------------------------------------
__global__ void gemm_x_kernel(const float* __restrict__ accIn,
                              const float* __restrict__ accOut,
                              const float* __restrict__ embed,
                              const float* __restrict__ wIn,
                              const float* __restrict__ wOut,
                              const float* __restrict__ wLoop,
                              const float* __restrict__ loopW,
                              const float* __restrict__ bias,
                              const float* __restrict__ gamma,
                              const float* __restrict__ beta,
                              float* __restrict__ xOut) {
    __shared__ float sA[2][BM * 16];   // 160 rows x 16 K-floats
    __shared__ float sB[2][16 * BN];   // 16 K-rows x 128 N-floats

    const int tid  = threadIdx.x;
    const int lane = tid & 31;
    const int wv   = tid >> 5;                 // 0..9
    const int m0   = blockIdx.x * BM;
    const int n0   = blockIdx.y * BN;

    // WMMA lane mapping (16x16x4 f32)
    const int aM = lane & 15;
    const int k2 = (lane >> 4) * 2;
    const int bN = lane & 15;

    v8f acc[8];
#pragma unroll
    for (int nt = 0; nt < 8; ++nt)
        acc[nt] = (v8f){0.f, 0.f, 0.f, 0.f, 0.f, 0.f, 0.f, 0.f};

    // 40 issue slots per K-tile (4 per wave, uniform so s_wait_asynccnt 4 is
    // correct double-buffering for every wave):
    //   slots 0..19 : A tile, 8 rows x 16 floats each (b128 per lane)
    //   slots 20..35: B tile, 1 row x 128 floats each
    //   slots 36..39: benign re-issue of B rows 0..3 (same data)
    auto issue_tile = [&](int t, int buf) {
        int src = t >> 4;
        int k0  = (t & 15) << 4;
        const float* Ag = (src == 0) ? accIn : (src == 1) ? accOut : embed;
        const float* Wg = (src == 0) ? wIn   : (src == 1) ? wOut   : wLoop;
#pragma unroll
        for (int i = 0; i < 4; ++i) {
            int slot = wv * 4 + i;
            if (slot < 20) {
                int row = slot * 8 + (lane >> 2);       // 0..159
                int q   = (lane & 3) * 4;               // float offset in row
                const float* gp = Ag + (size_t)(m0 + row) * DIM + k0 + q;
                float* lp = &sA[buf][row * 16 + q];
#if HAVE_ASYNC_LDS
                ASYNC_LD_B128(gp, lp);
#else
                *(float4*)lp = *(const float4*)gp;
#endif
            } else {
                int bs = slot - 20;
                if (bs >= 16) bs -= 16;                 // re-issue rows 0..3
                const float* gp = Wg + (size_t)(k0 + bs) * DIM + n0 + lane * 4;
                float* lp = &sB[buf][bs * BN + lane * 4];
#if HAVE_ASYNC_LDS
                ASYNC_LD_B128(gp, lp);
#else
                *(float4*)lp = *(const float4*)gp;
#endif
            }
        }
    };

    issue_tile(0, 0);

    for (int t = 0; t < NT_STEPS; ++t) {
        int buf = t & 1;
        if (t + 1 < NT_STEPS) {
            issue_tile(t + 1, buf ^ 1);
            WAIT_ASYNC(4);             // own 4 loads for tile t complete
        } else {
            WAIT_ASYNC(0);
        }
        __syncthreads();               // everyone's tile-t data in LDS

#pragma unroll
        for (int kk = 0; kk < 16; kk += 4) {
            v2f a;
            a.x = sA[buf][(wv * 16 + aM) * 16 + kk + k2];
            a.y = sA[buf][(wv * 16 + aM) * 16 + kk + k2 + 1];
#pragma unroll
            for (int nt = 0; nt < 8; ++nt) {
                v2f b;
                b.x = sB[buf][(kk + k2) * BN + nt * 16 + bN];
                b.y = sB[buf][(kk + k2 + 1) * BN + nt * 16 + bN];
                acc[nt] = __builtin_amdgcn_wmma_f32_16x16x4_f32(
                              false, a, false, b, (short)0, acc[nt],
                              false, false);
            }
        }
        __syncthreads();               // reads done before buf is overwritten
    }

    // epilogue: VGPR v of acc[nt] holds row = wv*16 + v + 8*(lane>=16),
    // col = nt*16 + (lane&15)
    const float invBn = rsqrtf(1.0f + 1e-5f);
    const int rBase = m0 + wv * 16 + ((lane >> 4) << 3);
#pragma unroll
    for (int nt = 0; nt < 8; ++nt) {
        int col = n0 + nt * 16 + (lane & 15);
        float g  = gamma[col] * invBn;
        float bb = beta[col];
        float cb = bias[col];
        float lw = loopW[col];
#pragma unroll
        for (int v = 0; v < 8; ++v) {
            float o = (acc[nt][v] - lw) * (1.0f / 3.0f) + cb;
            o = o * g + bb;
            xOut[(size_t)(rBase + v) * DIM + col] = tanhf(o);
        }
    }
}

// ---------------------------------------------------------------------------
// 6) rel_out = (rel_embed @ w_rel)[:400]   (small WMMA GEMM, M padded to 416)
// ---------------------------------------------------------------------------
__global__ void gemm_rel_kernel(const float* __restrict__ relE,
                                const float* __restrict__ wRel,
                                float* __restrict__ relOut) {
    int gid  = blockIdx.x * blockDim.x + threadIdx.x;
    int lane = gid & 31;
    int wv   = gid >> 5;
    const int ntiles = DIM / 16;
    int m0 = (wv / ntiles) * 16;
    int n0 = (wv % ntiles) * 16;
    if (m0 >= REL_ROWS) return;

    int aM = lane & 15;
    int k2 = (lane >> 4) * 2;
    int bN = lane & 15;
    int row = m0 + aM;
    bool inA = row < REL_ROWS;

    v8f c = {0.f, 0.f, 0.f, 0.f, 0.f, 0.f, 0.f, 0.f};
    for (int k = 0; k < DIM; k += 4) {
        v2f a, b;
        a.x = inA ? relE[(size_t)row * DIM + k + k2]     : 0.0f;
        a.y = inA ? relE[(size_t)row * DIM + k + k2 + 1] : 0.0f;
        b.x = wRel[(size_t)(k + k2) * DIM + n0 + bN];
        b.y = wRel[(size_t)(k + k2 + 1) * DIM + n0 + bN];
        c = __builtin_amdgcn_wmma_f32_16x16x4_f32(
                false, a, false, b, (short)0, c, false, false);
    }
    int nCol  = n0 + (lane & 15);
    int mBase = m0 + ((lane >> 4) << 3);
#pragma unroll
    for (int v = 0; v < 8; ++v) {
        int r = mBase + v;
        if (r < REL_OUT_ROWS) relOut[(size_t)r * DIM + nCol] = c[v];
    }
}

// ---------------------------------------------------------------------------
// 7) scoring: one wave32 per sample, ||x[h] + rel[r] - x[t] + eps||_2
// ---------------------------------------------------------------------------
__global__ void score_kernel(const float* __restrict__ x,
                             const float* __restrict__ relOut,
                             const int* __restrict__ sample,
                             float* __restrict__ out) {
    int gid  = blockIdx.x * blockDim.x + threadIdx.x;
    int s    = gid >> 5;
    int lane = gid & 31;
    if (s >= BATCH) return;
    int h = sample[s * 3 + 0];
    int r = sample[s * 3 + 1];
    int t = sample[s * 3 + 2];
    const float* xh = x      + (size_t)h * DIM;
    const float* rr = relOut + (size_t)r * DIM;
    const float* xt = x      + (size_t)t * DIM;
    float acc = 0.0f;
#pragma unroll
    for (int j = 0; j < 8; ++j) {
        int d = lane + 32 * j;
        float diff = xh[d] + rr[d] - xt[d] + 1e-6f;
        acc += diff * diff;
    }
#pragma unroll
    for (int off = 16; off > 0; off >>= 1)
        acc += __shfl_xor(acc, off, 32);
    if (lane == 0) out[s] = sqrtf(acc);
}

// ---------------------------------------------------------------------------
extern "C" void kernel_launch(void* const* d_in, const int* in_sizes, int n_in,
                              void* d_out, int out_size, void* d_ws, size_t ws_size,
                              hipStream_t stream) {
    const float* init_embed = (const float*)d_in[0];
    const float* init_rel   = (const float*)d_in[1];
    const float* loop_rel   = (const float*)d_in[2];
    const float* w_in       = (const float*)d_in[3];
    const float* w_out      = (const float*)d_in[4];
    const float* w_loop     = (const float*)d_in[5];
    const float* w_rel      = (const float*)d_in[6];
    const float* conv_bias  = (const float*)d_in[7];
    const float* bn_gamma   = (const float*)d_in[8];
    const float* bn_beta    = (const float*)d_in[9];
    const int*   edge_index = (const int*)d_in[10];
    const int*   edge_type  = (const int*)d_in[11];
    const int*   sample     = (const int*)d_in[12];
    float*       out        = (float*)d_out;

    // workspace: [acc_in | acc_out | deg_in | deg_out | x | relE | relO | loopW]
    float* acc_in  = (float*)d_ws;
    float* acc_out = acc_in  + (size_t)N_ENT * DIM;
    float* deg_in  = acc_out + (size_t)N_ENT * DIM;
    float* deg_out = deg_in  + N_ENT;
    float* xbuf    = deg_out + N_ENT;
    float* relE    = xbuf    + (size_t)N_ENT * DIM;
    float* relO    = relE    + (size_t)REL_ROWS * DIM;
    float* loopW   = relO    + (size_t)REL_OUT_ROWS * DIM;

    // zero acc_in, acc_out, deg_in, deg_out (contiguous)
    size_t zero_bytes = ((size_t)2 * N_ENT * DIM + 2 * N_ENT) * sizeof(float);
    (void)hipMemsetAsync(acc_in, 0, zero_bytes, stream);

    deg_kernel<<<(N_EDGE + 255) / 256, 256, 0, stream>>>(edge_index, deg_in, deg_out);

    build_rel_kernel<<<(REL_ROWS * DIM + 255) / 256, 256, 0, stream>>>(
        init_rel, loop_rel, relE);

    loopw_kernel<<<1, 256, 0, stream>>>(loop_rel, w_loop, loopW);

    {
        long long threads = (long long)HALF_E * 32;
        int blocks = (int)((threads + 255) / 256);
        agg_kernel<<<blocks, 256, 0, stream>>>(edge_index, edge_type, init_embed,
                                               relE, deg_in, acc_in, 0);
        agg_kernel<<<blocks, 256, 0, stream>>>(edge_index, edge_type, init_embed,
                                               relE, deg_out, acc_out, HALF_E);
    }

    {
        dim3 grid(N_ENT / BM, DIM / BN);           // 625 x 2
        gemm_x_kernel<<<grid, 32 * NWAVES, 0, stream>>>(
            acc_in, acc_out, init_embed, w_in, w_out, w_loop, loopW,
            conv_bias, bn_gamma, bn_beta, xbuf);
    }

    {
        int mtiles = (REL_ROWS + 15) / 16;         // 26
        int waves  = mtiles * (DIM / 16);          // 416
        int blocks = (waves * 32 + 255) / 256;     // 52
        gemm_rel_kernel<<<blocks, 256, 0, stream>>>(relE, w_rel, relO);
    }

    {
        int blocks = (BATCH * 32 + 255) / 256;     // 512
        score_kernel<<<blocks, 256, 0, stream>>>(xbuf, relO, sample, out);
    }
}